// GetDensity_13932873908299
// MI455X (gfx1250) — compile-verified
//
#include <hip/hip_runtime.h>
#include <math.h>

typedef float v2f __attribute__((ext_vector_type(2)));
typedef float v8f __attribute__((ext_vector_type(8)));

#define B_      8
#define N_      1000
#define P_      40000
#define E_      (B_*P_)        // 320000 edges; 320000 % 256 == 0
#define NW      8
#define TOT     (B_*N_)        // 8000 atoms
#define SUMW_N  (TOT*13*NW)    // 832000 f32 in d_ws (3.33 MB, L2 resident)

__global__ __launch_bounds__(256) void zero_kernel(float* ws) {
    int i = blockIdx.x * 256 + threadIdx.x;
    if (i < SUMW_N) ws[i] = 0.0f;
}

// One edge per lane. Wave = 32 edges = two 16-row WMMA tiles.
// G[edge, p, m] = sum_k (fc*radial_k) * hyper[p][k][m] via V_WMMA_F32_16X16X4_F32:
//   A (16x4): M = edge row, K = radial index (K=8 -> 2 chunks of 4)
//   B (4x16): K = radial index, N cols 0-7 -> order0 m, 8-15 -> order1 m (matrix "p01");
//             second matrix "p2": cols 0-7 -> order2 m, cols 8-15 zero-padded.
__global__ __launch_bounds__(256) void edge_kernel(
    const float* __restrict__ cart,   const int* __restrict__ species,
    const int*   __restrict__ aidx,   const float* __restrict__ shifts,
    const float* __restrict__ rs,     const float* __restrict__ inta,
    const float* __restrict__ params, const float* __restrict__ hyper,
    float* __restrict__ sumw)
{
    __shared__ float lds_r[256 * NW];          // fc*radial per thread      (8 KB)
    __shared__ float lds_g[8][2][16][24];      // G per wave/tile/row       (24 KB)

    const int tid   = threadIdx.x;
    const int lane  = tid & 31;
    const int wave  = tid >> 5;
    const int nco   = lane & 15;               // N column / M row within tile
    const int khalf = (lane >> 4) & 1;         // 0: K{0,1}, 1: K{2,3} of a chunk
    const int e     = blockIdx.x * 256 + tid;

    // ---- constant B fragments (hyper is 3x8x8 f32, broadcast loads) ----
    // B 4x16 layout assumed analogous to C/D: vgpr0 -> K = khalf*2, vgpr1 -> K = khalf*2+1
    v2f b01[2], b2[2];
    {
        const int p01 = nco >> 3, m = nco & 7;
        const float z = (nco < 8) ? 1.0f : 0.0f;
        #pragma unroll
        for (int c = 0; c < 2; ++c) {
            const int k0 = c * 4 + khalf * 2;
            b01[c].x = hyper[p01 * 64 + (k0 + 0) * 8 + m];
            b01[c].y = hyper[p01 * 64 + (k0 + 1) * 8 + m];
            b2 [c].x = z * hyper[2 * 64 + (k0 + 0) * 8 + m];
            b2 [c].y = z * hyper[2 * 64 + (k0 + 1) * 8 + m];
        }
    }

    // ---- per-edge geometry ----
    const int b  = e / P_;
    const int i0 = aidx[e]      + b * N_;      // center atom (flat)
    const int i1 = aidx[E_ + e] + b * N_;      // neighbor atom (flat)
    const float dx = cart[i0*3+0] - cart[i1*3+0] + shifts[e*3+0];
    const float dy = cart[i0*3+1] - cart[i1*3+1] + shifts[e*3+1];
    const float dz = cart[i0*3+2] - cart[i1*3+2] + shifts[e*3+2];
    const float d2 = dx*dx + dy*dy + dz*dz;
    // dist = sqrt(d2), u = dvec/dist: fuse both through one rsqrt (d2 > 0 by harness)
    const float inv  = __frsqrt_rn(d2);
    const float dist = d2 * inv;
    const float ux = dx*inv, uy = dy*inv, uz = dz*inv;
    const int   sp = species[i1];

    const float c0 = 0.5f * __cosf(dist * 0.6283185307f) + 0.5f;   // pi/CUTOFF
    const float fc = c0 * c0;

    #pragma unroll
    for (int k = 0; k < NW; ++k) {
        const float d = dist - rs[sp * NW + k];
        lds_r[tid * NW + k] = fc * __expf(-inta[sp * NW + k] * d * d);
    }
    __syncthreads();

    // ---- WMMA: two 16-edge tiles, K=8 as 2 chunks, two B matrices ----
    // A 16x4 ISA layout: lanes 0-15 row=lane K0/K1 in vgpr0/1; lanes 16-31 K2/K3.
    const float* rw = &lds_r[wave * 32 * NW];
    v8f d01[2] = {{0,0,0,0,0,0,0,0},{0,0,0,0,0,0,0,0}};
    v8f d2v[2] = {{0,0,0,0,0,0,0,0},{0,0,0,0,0,0,0,0}};
    #pragma unroll
    for (int t = 0; t < 2; ++t) {
        #pragma unroll
        for (int c = 0; c < 2; ++c) {
            const float* ap = &rw[(t * 16 + nco) * NW + c * 4 + khalf * 2];
            v2f a; a.x = ap[0]; a.y = ap[1];
            d01[t] = __builtin_amdgcn_wmma_f32_16x16x4_f32(
                false, a, false, b01[c], (short)0, d01[t], false, false);
            d2v[t] = __builtin_amdgcn_wmma_f32_16x16x4_f32(
                false, a, false, b2[c],  (short)0, d2v[t], false, false);
        }
    }

    // ---- D (16x16 C layout: vgpr v -> row v + khalf*8, col = nco) -> LDS ----
    #pragma unroll
    for (int t = 0; t < 2; ++t)
        #pragma unroll
        for (int v = 0; v < 8; ++v)
            lds_g[wave][t][v + khalf * 8][nco] = d01[t][v];
    if (nco < 8) {                              // divergence only after all WMMAs
        #pragma unroll
        for (int t = 0; t < 2; ++t)
            #pragma unroll
            for (int v = 0; v < 8; ++v)
                lds_g[wave][t][v + khalf * 8][16 + nco] = d2v[t][v];
    }
    __syncthreads();

    // ---- gather this edge's 24 G values, fold in per-species coeff ----
    const int mytile = lane >> 4, myrow = lane & 15;
    const float* co = &params[sp * 24];
    float h[24];
    #pragma unroll
    for (int q = 0; q < 24; ++q)
        h[q] = lds_g[wave][mytile][myrow][q] * co[q];

    float ang[13];
    ang[0] = 1.0f; ang[1] = ux;   ang[2] = uy;   ang[3] = uz;
    ang[4] = ux*ux; ang[5] = ux*uy; ang[6] = ux*uz;
    ang[7] = uy*ux; ang[8] = uy*uy; ang[9] = uy*uz;
    ang[10]= uz*ux; ang[11]= uz*uy; ang[12]= uz*uz;

    // ---- scatter-add into sumw[center][13][8] (L2-resident f32 atomics) ----
    float* base = sumw + i0 * (13 * NW);
    #pragma unroll
    for (int m = 0; m < NW; ++m) atomicAdd(base + m, h[m]);
    #pragma unroll
    for (int j = 1; j < 4; ++j)
        #pragma unroll
        for (int m = 0; m < NW; ++m) atomicAdd(base + j*NW + m, ang[j] * h[8 + m]);
    #pragma unroll
    for (int j = 4; j < 13; ++j)
        #pragma unroll
        for (int m = 0; m < NW; ++m) atomicAdd(base + j*NW + m, ang[j] * h[16 + m]);
}

// density[t][p][m] = sum_{j in order p} sumw[t][j][m]^2 ; one thread per (atom, m)
__global__ __launch_bounds__(256) void reduce_kernel(const float* __restrict__ sumw,
                                                     float* __restrict__ out) {
    int t = blockIdx.x * 256 + threadIdx.x;
    if (t >= TOT * NW) return;
    const int atom = t >> 3, m = t & 7;
    const float* s = sumw + atom * (13 * NW) + m;
    float d0 = s[0] * s[0], d1 = 0.0f, d2 = 0.0f;
    #pragma unroll
    for (int j = 1; j < 4; ++j)  { float x = s[j * NW]; d1 += x * x; }
    #pragma unroll
    for (int j = 4; j < 13; ++j) { float x = s[j * NW]; d2 += x * x; }
    out[atom * 24 +      m] = d0;
    out[atom * 24 +  8 + m] = d1;
    out[atom * 24 + 16 + m] = d2;
}

extern "C" void kernel_launch(void* const* d_in, const int* in_sizes, int n_in,
                              void* d_out, int out_size, void* d_ws, size_t ws_size,
                              hipStream_t stream) {
    (void)in_sizes; (void)n_in; (void)out_size; (void)ws_size;
    const float* cart    = (const float*)d_in[0];
    /* d_in[1] numatoms: all molecules full, unused */
    const int*   species = (const int*)d_in[2];
    const int*   aidx    = (const int*)d_in[3];
    const float* shifts  = (const float*)d_in[4];
    const float* rs      = (const float*)d_in[5];
    const float* inta    = (const float*)d_in[6];
    const float* params  = (const float*)d_in[7];
    const float* hyper   = (const float*)d_in[8];
    float* out  = (float*)d_out;
    float* sumw = (float*)d_ws;

    zero_kernel  <<<(SUMW_N + 255) / 256, 256, 0, stream>>>(sumw);
    edge_kernel  <<<E_ / 256,             256, 0, stream>>>(cart, species, aidx, shifts,
                                                            rs, inta, params, hyper, sumw);
    reduce_kernel<<<(TOT * NW + 255)/256, 256, 0, stream>>>(sumw, out);
}